// OT_loss_57458072486124
// MI455X (gfx1250) — compile-verified
//
#include <hip/hip_runtime.h>
#include <hip/hip_bf16.h>
#include <math.h>

// Shapes fixed by the reference: B=1024, V=50000, D=256.
#define NROWS   1024
#define NV      50000
#define NV4     12500          // 50000 / 4
#define ND      256
#define N_BD4   65536          // 1024*256/4
#define NB2     64             // blocks for KLD/Bures kernel

typedef float v2f __attribute__((ext_vector_type(2)));
typedef float v8f __attribute__((ext_vector_type(8)));

// ---------------------------------------------------------------------------
// Wave-wide (32-lane) sum via V_WMMA_F32_16X16X4_F32.
// A (16x4) layout (ISA 7.12.2): lanes0-15 {VGPR0->K0, VGPR1->K1},
// lanes16-31 {VGPR0->K2, VGPR1->K3}.  With a = (v, 0) and B = all-ones:
//   D[m][n] = v[m] + v[m+16]   (all-ones B is layout invariant)
// C/D layout: vgpr r -> row r (lanes 0-15) / row r+8 (lanes 16-31), so
// sum_r c[r] gives rows 0-7 resp. 8-15; one xor-16 shuffle finishes it.
// Requires EXEC all-ones: only call from full, non-divergent blocks.
// ---------------------------------------------------------------------------
__device__ __forceinline__ float wave_sum_wmma(float v) {
  v2f a; a[0] = v;    a[1] = 0.0f;
  v2f b; b[0] = 1.0f; b[1] = 1.0f;
  v8f c = {};
  c = __builtin_amdgcn_wmma_f32_16x16x4_f32(false, a, false, b,
                                            (short)0, c, false, false);
  float t = c[0] + c[1] + c[2] + c[3] + c[4] + c[5] + c[6] + c[7];
  t += __shfl_xor(t, 16, 32);
  return t;   // full 32-lane sum, valid in every lane
}

// Block(256-thread)-wide sum: WMMA wave sums + 8-entry LDS combine.
__device__ __forceinline__ float block_sum_256(float v, float* l8) {
  float t = wave_sum_wmma(v);
  __syncthreads();                       // protect l8 reuse across calls
  if ((threadIdx.x & 31) == 0) l8[threadIdx.x >> 5] = t;
  __syncthreads();
  return l8[0] + l8[1] + l8[2] + l8[3] + l8[4] + l8[5] + l8[6] + l8[7];
}

// Online-softmax update for one element of one logit stream.
#define ONLINE(lv_, m_, s_, d_, xv_)                                   \
  do {                                                                 \
    float _l = (lv_);                                                  \
    if (_l > m_) { s_ = s_ * __expf(m_ - _l) + 1.0f; m_ = _l; }        \
    else         { s_ += __expf(_l - m_); }                            \
    d_ = fmaf(_l, (xv_), d_);                                          \
  } while (0)

// ---------------------------------------------------------------------------
// Kernel A: one block per row; streams x + 3 logit rows once (float4),
// keeps per-thread online-softmax state for all 3 streams, then reduces
// with WMMA wave sums + LDS and writes 3 per-row terms:
//   rowterm_q = dot(l_q, x) - (M_q + log S_q) * sum(x)
// ---------------------------------------------------------------------------
__global__ __launch_bounds__(512) void bce_rows_kernel(
    const float4* __restrict__ g0,   // recon_x
    const float4* __restrict__ g1,   // logits_text
    const float4* __restrict__ g2,   // logits_rec
    const float4* __restrict__ gx,   // x
    float* __restrict__ rowterm)     // [3 * NROWS]
{
  const int row = blockIdx.x;
  const size_t base = (size_t)row * NV4;
  const float4* a0 = g0 + base;
  const float4* a1 = g1 + base;
  const float4* a2 = g2 + base;
  const float4* ax = gx + base;

  const float NEG_INF = -__builtin_inff();
  float m0 = NEG_INF, m1 = NEG_INF, m2 = NEG_INF;
  float s0 = 0.f, s1 = 0.f, s2 = 0.f;
  float d0 = 0.f, d1 = 0.f, d2 = 0.f;
  float sx = 0.f;

  for (int i = threadIdx.x; i < NV4; i += 512) {
    float4 xv = ax[i];
    float4 A  = a0[i];
    float4 Bv = a1[i];
    float4 Cv = a2[i];
    sx += xv.x + xv.y + xv.z + xv.w;
    ONLINE(A.x,  m0, s0, d0, xv.x); ONLINE(A.y,  m0, s0, d0, xv.y);
    ONLINE(A.z,  m0, s0, d0, xv.z); ONLINE(A.w,  m0, s0, d0, xv.w);
    ONLINE(Bv.x, m1, s1, d1, xv.x); ONLINE(Bv.y, m1, s1, d1, xv.y);
    ONLINE(Bv.z, m1, s1, d1, xv.z); ONLINE(Bv.w, m1, s1, d1, xv.w);
    ONLINE(Cv.x, m2, s2, d2, xv.x); ONLINE(Cv.y, m2, s2, d2, xv.y);
    ONLINE(Cv.z, m2, s2, d2, xv.z); ONLINE(Cv.w, m2, s2, d2, xv.w);
  }

  // ---- wave-level max of the running maxima (butterfly) ----
  float M0 = m0, M1 = m1, M2 = m2;
  #pragma unroll
  for (int off = 16; off > 0; off >>= 1) {
    M0 = fmaxf(M0, __shfl_xor(M0, off, 32));
    M1 = fmaxf(M1, __shfl_xor(M1, off, 32));
    M2 = fmaxf(M2, __shfl_xor(M2, off, 32));
  }
  // rescale each lane's exp-sum to the wave max
  s0 *= __expf(m0 - M0);
  s1 *= __expf(m1 - M1);
  s2 *= __expf(m2 - M2);

  // ---- wave-level sums via WMMA (7 quantities) ----
  float wSX = wave_sum_wmma(sx);
  float wD0 = wave_sum_wmma(d0);
  float wD1 = wave_sum_wmma(d1);
  float wD2 = wave_sum_wmma(d2);
  float wS0 = wave_sum_wmma(s0);
  float wS1 = wave_sum_wmma(s1);
  float wS2 = wave_sum_wmma(s2);

  // ---- cross-wave combine (16 waves) ----
  __shared__ float Ls[7][16];
  __shared__ float Lm[3][16];
  const int lane = threadIdx.x & 31;
  const int wv   = threadIdx.x >> 5;
  if (lane == 0) {
    Ls[0][wv] = wSX;
    Ls[1][wv] = wD0; Ls[2][wv] = wD1; Ls[3][wv] = wD2;
    Ls[4][wv] = wS0; Ls[5][wv] = wS1; Ls[6][wv] = wS2;
    Lm[0][wv] = M0;  Lm[1][wv] = M1;  Lm[2][wv] = M2;
  }
  __syncthreads();

  if (threadIdx.x == 0) {
    float B0 = NEG_INF, B1 = NEG_INF, B2 = NEG_INF;
    #pragma unroll
    for (int w = 0; w < 16; ++w) {
      B0 = fmaxf(B0, Lm[0][w]);
      B1 = fmaxf(B1, Lm[1][w]);
      B2 = fmaxf(B2, Lm[2][w]);
    }
    float SX = 0.f, D0s = 0.f, D1s = 0.f, D2s = 0.f;
    float S0s = 0.f, S1s = 0.f, S2s = 0.f;
    #pragma unroll
    for (int w = 0; w < 16; ++w) {
      SX  += Ls[0][w];
      D0s += Ls[1][w]; D1s += Ls[2][w]; D2s += Ls[3][w];
      S0s += Ls[4][w] * __expf(Lm[0][w] - B0);
      S1s += Ls[5][w] * __expf(Lm[1][w] - B1);
      S2s += Ls[6][w] * __expf(Lm[2][w] - B2);
    }
    rowterm[0 * NROWS + row] = D0s - (B0 + logf(S0s)) * SX;
    rowterm[1 * NROWS + row] = D1s - (B1 + logf(S1s)) * SX;
    rowterm[2 * NROWS + row] = D2s - (B2 + logf(S2s)) * SX;
  }
}

// ---------------------------------------------------------------------------
// Kernel B: KLD + diagonal-Bures element sums over (1024, 256).
//   kld  elem: 1 + lv - mu^2 - e^lv
//   wass elem: (mu - rec_mu)^2 + (e^{lv/2} - e^{rlv/2})^2
// ---------------------------------------------------------------------------
__global__ __launch_bounds__(256) void kld_wass_kernel(
    const float4* __restrict__ gmu, const float4* __restrict__ glv,
    const float4* __restrict__ grmu, const float4* __restrict__ grlv,
    float* __restrict__ kld_part, float* __restrict__ wass_part)
{
  float k = 0.f, w = 0.f;
  const int stride = NB2 * 256;
  for (int i = blockIdx.x * 256 + threadIdx.x; i < N_BD4; i += stride) {
    float4 m  = gmu[i];
    float4 l  = glv[i];
    float4 rm = grmu[i];
    float4 rl = grlv[i];
    #define ELEM(mm, ll, rmm, rll)                            \
      {                                                       \
        float h1 = __expf(0.5f * (ll));                       \
        float h2 = __expf(0.5f * (rll));                      \
        float e1 = h1 * h1;                                   \
        k += 1.0f + (ll) - (mm) * (mm) - e1;                  \
        float dm = (mm) - (rmm);                              \
        float dh = h1 - h2;                                   \
        w += dm * dm + dh * dh;                               \
      }
    ELEM(m.x, l.x, rm.x, rl.x); ELEM(m.y, l.y, rm.y, rl.y);
    ELEM(m.z, l.z, rm.z, rl.z); ELEM(m.w, l.w, rm.w, rl.w);
    #undef ELEM
  }
  __shared__ float l8[8];
  float kt = block_sum_256(k, l8);
  float wt = block_sum_256(w, l8);
  if (threadIdx.x == 0) {
    kld_part[blockIdx.x]  = kt;
    wass_part[blockIdx.x] = wt;
  }
}

// ---------------------------------------------------------------------------
// Kernel C: deterministic finalize. Reduces per-row / per-block partials,
// assembles the 6 outputs in reference return order.
// ---------------------------------------------------------------------------
__global__ __launch_bounds__(256) void finalize_kernel(
    const float* __restrict__ rowterm,
    const float* __restrict__ kld_part,
    const float* __restrict__ wass_part,
    const int* __restrict__ anneal_p,
    const int* __restrict__ eps_p,
    float* __restrict__ out)
{
  __shared__ float l8[8];
  float a0 = 0.f, a1 = 0.f, a2 = 0.f;
  for (int r = threadIdx.x; r < NROWS; r += 256) {
    a0 += rowterm[0 * NROWS + r];
    a1 += rowterm[1 * NROWS + r];
    a2 += rowterm[2 * NROWS + r];
  }
  float t0 = block_sum_256(a0, l8);
  float t1 = block_sum_256(a1, l8);
  float t2 = block_sum_256(a2, l8);

  float kv = (threadIdx.x < NB2) ? kld_part[threadIdx.x]  : 0.f;
  float wv = (threadIdx.x < NB2) ? wass_part[threadIdx.x] : 0.f;
  float kt = block_sum_256(kv, l8);
  float wt = block_sum_256(wv, l8);

  if (threadIdx.x == 0) {
    const float invBV = 1.0f / ((float)NROWS * (float)NV);
    float BCE_merged = -t0 * invBV;
    float BCE_text   = -t1 * invBV;
    float BCE_rec    = -t2 * invBV;
    float BCE  = (BCE_merged + BCE_text + BCE_rec) * (1.0f / 3.0f);  // taus = 1
    float KLD  = -0.5f * kt / ((float)NROWS * (float)ND);
    float WASS = wt / (float)NROWS;
    float L = BCE + (float)anneal_p[0] * KLD + (float)eps_p[0] * WASS;
    out[0] = L;
    out[1] = BCE;
    out[2] = WASS;
    out[3] = BCE_rec;
    out[4] = BCE_text;
    out[5] = BCE_merged;
  }
}

// ---------------------------------------------------------------------------
extern "C" void kernel_launch(void* const* d_in, const int* in_sizes, int n_in,
                              void* d_out, int out_size, void* d_ws, size_t ws_size,
                              hipStream_t stream) {
  const float4* recon = (const float4*)d_in[0];
  const float4* ltext = (const float4*)d_in[1];
  const float4* lrec  = (const float4*)d_in[2];
  const float4* x     = (const float4*)d_in[3];
  const float4* mu    = (const float4*)d_in[4];
  const float4* lv    = (const float4*)d_in[5];
  const float4* rmu   = (const float4*)d_in[6];
  const float4* rlv   = (const float4*)d_in[7];
  const int* anneal   = (const int*)d_in[8];
  const int* epsilon  = (const int*)d_in[9];

  float* wsf      = (float*)d_ws;
  float* rowterm  = wsf;                 // 3 * 1024 floats
  float* kld_p    = wsf + 3 * NROWS;     // 64 floats
  float* wass_p   = kld_p + NB2;         // 64 floats

  bce_rows_kernel<<<NROWS, 512, 0, stream>>>(recon, ltext, lrec, x, rowterm);
  kld_wass_kernel<<<NB2, 256, 0, stream>>>(mu, lv, rmu, rlv, kld_p, wass_p);
  finalize_kernel<<<1, 256, 0, stream>>>(rowterm, kld_p, wass_p,
                                         anneal, epsilon, (float*)d_out);
}